// NormBlock_31542239822284
// MI455X (gfx1250) — compile-verified
//
#include <hip/hip_runtime.h>
#include <hip/hip_bf16.h>

// x: (32, 1, 8192, 256) fp32. MaxPool k=(3,1) s=(2,1) pad=(1,0) along H
// -> (32, 1, 4096, 256), then per-sample min-max normalize.
#define BATCH 32
#define H_IN  8192
#define H_OUT 4096
#define W     256
#define W4    64                      // vec4 columns per row
#define OROWS 32                      // output rows per block
#define IROWS (2 * OROWS + 1)         // 65 input rows staged in LDS
#define NCHUNK  ((IROWS * W * 4) / 16)        // 4160 x 16B chunks (m > 0)
#define NCHUNK0 (((IROWS - 1) * W * 4) / 16)  // 4096 x 16B chunks (m == 0)

typedef float f32x4 __attribute__((ext_vector_type(4)));

// ---------- order-preserving float <-> uint encoding (for atomic min/max) ----
__device__ __forceinline__ unsigned enc_f32(float f) {
    unsigned u = __float_as_uint(f);
    return (u & 0x80000000u) ? ~u : (u | 0x80000000u);
}
__device__ __forceinline__ float dec_f32(unsigned e) {
    unsigned u = (e & 0x80000000u) ? (e ^ 0x80000000u) : ~e;
    return __uint_as_float(u);
}

__device__ __forceinline__ f32x4 vmax4(f32x4 a, f32x4 b) {
    f32x4 r;
    r.x = fmaxf(a.x, b.x); r.y = fmaxf(a.y, b.y);
    r.z = fmaxf(a.z, b.z); r.w = fmaxf(a.w, b.w);
    return r;
}

// ---------------------------------------------------------------------------
// Kernel A: init per-sample encoded (min,max) slots in workspace.
// ---------------------------------------------------------------------------
__global__ void nb_init_minmax(unsigned* __restrict__ mm) {
    int i = threadIdx.x;
    if (i < BATCH) {
        mm[2 * i + 0] = 0xFFFFFFFFu;   // min slot (largest encoded)
        mm[2 * i + 1] = 0x00000000u;   // max slot (smallest encoded)
    }
}

// ---------------------------------------------------------------------------
// Kernel B: async-stage a 65-row input strip into LDS, maxpool from LDS,
// store 32 pooled rows, and reduce per-sample min/max.
// grid = (128, 32), block = 256 (8 wave32 waves). LDS = 65 KiB/block.
// ---------------------------------------------------------------------------
__global__ __launch_bounds__(256) void nb_pool_minmax(
    const float* __restrict__ x, float* __restrict__ out,
    unsigned* __restrict__ mm) {

    __shared__ float smem[IROWS * W];        // 65 rows x 1 KiB
    __shared__ unsigned smin, smax;

    const int b = blockIdx.y;
    const int m = blockIdx.x;                // strip index: output rows 32m..32m+31

    // Strip setup. LDS slot s holds input row (r_start + s); for m == 0 the
    // virtual row -1 (maxpool -inf padding) occupies slot 0.
    const float* gbase;
    unsigned lds_start;
    int nchunks;
    if (m == 0) {
        gbase = x + (size_t)b * H_IN * W;                      // rows 0..63
        lds_start = W * 4;                                     // into slot 1
        nchunks = NCHUNK0;
    } else {
        gbase = x + ((size_t)b * H_IN + (64 * m - 1)) * W;     // rows 64m-1..64m+63
        lds_start = 0;
        nchunks = NCHUNK;
    }

    if (threadIdx.x == 0) { smin = 0xFFFFFFFFu; smax = 0u; }
    if (m == 0 && threadIdx.x < 64) {        // slot 0 = -inf padding row
        const float ninf = -__builtin_inff();
        f32x4 v; v.x = ninf; v.y = ninf; v.z = ninf; v.w = ninf;
        *(f32x4*)&smem[threadIdx.x * 4] = v;
    }

    // Async copy global -> LDS in 16B chunks (each wave instruction moves
    // 32 lanes x 16B = 512B; tracked on ASYNCcnt, no VGPR landing zone).
    const unsigned lds0 = (unsigned)(size_t)&smem[0] + lds_start;
    const unsigned long long g0 = (unsigned long long)(size_t)gbase;
    for (int ci = threadIdx.x; ci < nchunks; ci += 256) {
        unsigned           l = lds0 + (unsigned)ci * 16u;
        unsigned long long g = g0 + (unsigned long long)ci * 16u;
        asm volatile("global_load_async_to_lds_b128 %0, %1, off"
                     :: "v"(l), "v"(g) : "memory");
    }
    asm volatile("s_wait_asynccnt 0x0" ::: "memory");
    __syncthreads();

    // Compute: thread t handles vec4 column (t & 63) of output rows
    // oh = (t >> 6) + 4j, j = 0..7. Window = LDS slots 2oh, 2oh+1, 2oh+2.
    const int wv = threadIdx.x & 63;
    const int g  = threadIdx.x >> 6;
    f32x4* __restrict__ o = (f32x4*)(out + (size_t)b * H_OUT * W);
    const int h0 = m * OROWS;

    float tmin = __builtin_inff();
    float tmax = -__builtin_inff();

    #pragma unroll
    for (int j = 0; j < 8; ++j) {
        const int oh = g + 4 * j;
        const f32x4 s0 = *(const f32x4*)&smem[(2 * oh + 0) * W + wv * 4];
        const f32x4 s1 = *(const f32x4*)&smem[(2 * oh + 1) * W + wv * 4];
        const f32x4 s2 = *(const f32x4*)&smem[(2 * oh + 2) * W + wv * 4];
        f32x4 r = vmax4(vmax4(s0, s1), s2);
        o[(size_t)(h0 + oh) * W4 + wv] = r;   // RT store: keep in L2 for pass 2
        tmax = fmaxf(tmax, fmaxf(fmaxf(r.x, r.y), fmaxf(r.z, r.w)));
        tmin = fminf(tmin, fminf(fminf(r.x, r.y), fminf(r.z, r.w)));
    }

    // Wave32 butterfly, then one LDS atomic pair per wave, one global pair
    // per block.
    #pragma unroll
    for (int off = 16; off > 0; off >>= 1) {
        tmin = fminf(tmin, __shfl_xor(tmin, off, 32));
        tmax = fmaxf(tmax, __shfl_xor(tmax, off, 32));
    }
    if ((threadIdx.x & 31) == 0) {
        atomicMin(&smin, enc_f32(tmin));
        atomicMax(&smax, enc_f32(tmax));
    }
    __syncthreads();
    if (threadIdx.x == 0) {
        atomicMin(&mm[2 * b + 0], smin);
        atomicMax(&mm[2 * b + 1], smax);
    }
}

// ---------------------------------------------------------------------------
// Kernel C: in-place per-sample normalization. Last use of these lines ->
// nontemporal (TH=NT) b128 loads/stores. grid = (1024, 32).
// ---------------------------------------------------------------------------
__global__ __launch_bounds__(256) void nb_normalize(
    float* __restrict__ out, const unsigned* __restrict__ mm) {

    const int b   = blockIdx.y;
    const int idx = blockIdx.x * 256 + threadIdx.x;   // 0 .. 262143

    const float mn = dec_f32(mm[2 * b + 0]);
    const float mx = dec_f32(mm[2 * b + 1]);
    const float d  = mx - mn;

    f32x4* __restrict__ o = (f32x4*)(out + (size_t)b * H_OUT * W);

    f32x4 v = __builtin_nontemporal_load(&o[idx]);
    v.x = (v.x - mn) / d;
    v.y = (v.y - mn) / d;
    v.z = (v.z - mn) / d;
    v.w = (v.w - mn) / d;
    __builtin_nontemporal_store(v, &o[idx]);
}

// ---------------------------------------------------------------------------
extern "C" void kernel_launch(void* const* d_in, const int* in_sizes, int n_in,
                              void* d_out, int out_size, void* d_ws,
                              size_t ws_size, hipStream_t stream) {
    const float* x   = (const float*)d_in[0];
    float*       out = (float*)d_out;
    unsigned*    mm  = (unsigned*)d_ws;   // 32 * 2 uints = 256 bytes

    nb_init_minmax<<<dim3(1), dim3(32), 0, stream>>>(mm);

    dim3 gridB(H_OUT / OROWS, BATCH, 1);  // (128, 32)
    nb_pool_minmax<<<gridB, 256, 0, stream>>>(x, out, mm);

    dim3 gridC(1024, BATCH, 1);
    nb_normalize<<<gridC, 256, 0, stream>>>(out, mm);
}